// QuantCat_40166534152675
// MI455X (gfx1250) — compile-verified
//
#include <hip/hip_runtime.h>

// LSQ fake-quant of concat([x1, x2], axis=1), x1,x2: [16, 768, 32, 32] fp32.
// Forward value: s_eff = max(s, 1e-5); out = clamp(rint(x/s_eff), -127, 127) * s_eff
//
// Memory-bound streaming kernel (roofline: ~201 MB @ 23.3 TB/s ~= 8.6 us):
//  - 128-bit global loads/stores with CDNA5 non-temporal TH hints (read/write-once
//    data, ~201 MB working set > 192 MB L2).
//  - Concat source resolved ONCE per block entirely on the SALU (chunk size is a
//    multiple of the block span), so loads are SGPR-base + VGPR-offset.
//  - v_med3_num_f32 clamp: 4 VALU/element (mul, rndne, med3, mul) -- minimal.
//  - 8 x float4 = 128 B per thread to amortize setup; wave32-sized 256-thread blocks.

typedef float v4f __attribute__((ext_vector_type(4)));

#define THD_POS 127.0f
#define THD_NEG (-127.0f)
#define EPS_Q 1e-5f

// One (batch, source) chunk of the concatenated output, in float4 units:
// 768 * 32 * 32 / 4 = 196608.
constexpr int kPerChunk4    = (768 * 32 * 32) / 4;
constexpr int kBlockThreads = 256;                      // 8 wave32 waves
constexpr int kUnroll       = 8;                        // 8 x float4 = 128 B / thread
constexpr int kBlockSpan    = kBlockThreads * kUnroll;  // 2048 float4 per block

static_assert(kPerChunk4 % kBlockSpan == 0,
              "chunk boundary must align to block span so the concat source is "
              "uniform (SALU-resolvable) per block");

__device__ __forceinline__ v4f fakequant4(v4f v, float inv_s, float s_eff)
{
    v4f r;
    r.x = __builtin_amdgcn_fmed3f(__builtin_rintf(v.x * inv_s), THD_NEG, THD_POS) * s_eff;
    r.y = __builtin_amdgcn_fmed3f(__builtin_rintf(v.y * inv_s), THD_NEG, THD_POS) * s_eff;
    r.z = __builtin_amdgcn_fmed3f(__builtin_rintf(v.z * inv_s), THD_NEG, THD_POS) * s_eff;
    r.w = __builtin_amdgcn_fmed3f(__builtin_rintf(v.w * inv_s), THD_NEG, THD_POS) * s_eff;
    return r;
}

__global__ __launch_bounds__(kBlockThreads)
void lsq_quantcat_kernel(const v4f* __restrict__ x1,
                         const v4f* __restrict__ x2,
                         const float* __restrict__ s,
                         v4f* __restrict__ out,
                         int n4)
{
    // Uniform scalar step size: one s_load; divide hoisted per-wave.
    const float s_eff = fmaxf(s[0], EPS_Q);
    const float inv_s = 1.0f / s_eff;

    // ---- All concat/source math is uniform (blockIdx only) -> SALU ----
    const int blockBase = (int)blockIdx.x * kBlockSpan;       // first float4 of block
    const int chunk     = blockBase / kPerChunk4;             // 0..31, alternates x1/x2
    const int within    = blockBase - chunk * kPerChunk4;
    const v4f* __restrict__ src = (chunk & 1) ? x2 : x1;      // s_cselect on pointer
    const int srcBase   = (chunk >> 1) * kPerChunk4 + within; // uniform source offset

    const int t = (int)threadIdx.x;

    if (blockBase + kBlockSpan <= n4) {
        // Fast path: whole block in range, no per-element guards.
#pragma unroll
        for (int k = 0; k < kUnroll; ++k) {
            const int off = t + k * kBlockThreads;
            v4f v = __builtin_nontemporal_load(&src[srcBase + off]);
            __builtin_nontemporal_store(fakequant4(v, inv_s, s_eff), &out[blockBase + off]);
        }
    } else {
        // Tail path (unused for the reference shapes, kept for generality).
#pragma unroll
        for (int k = 0; k < kUnroll; ++k) {
            const int off = t + k * kBlockThreads;
            if (blockBase + off < n4) {
                v4f v = __builtin_nontemporal_load(&src[srcBase + off]);
                __builtin_nontemporal_store(fakequant4(v, inv_s, s_eff), &out[blockBase + off]);
            }
        }
    }
}

extern "C" void kernel_launch(void* const* d_in, const int* in_sizes, int n_in,
                              void* d_out, int out_size, void* d_ws, size_t ws_size,
                              hipStream_t stream)
{
    const v4f*   x1  = reinterpret_cast<const v4f*>(d_in[0]);   // [16,768,32,32] f32
    const v4f*   x2  = reinterpret_cast<const v4f*>(d_in[1]);   // [16,768,32,32] f32
    const float* s   = reinterpret_cast<const float*>(d_in[2]); // [1] f32
    v4f*         out = reinterpret_cast<v4f*>(d_out);           // [16,1536,32,32] f32

    const int n4   = out_size / 4;                          // 6,291,456 float4
    const int grid = (n4 + kBlockSpan - 1) / kBlockSpan;    // 3072 blocks

    lsq_quantcat_kernel<<<grid, kBlockThreads, 0, stream>>>(x1, x2, s, out, n4);
}